// RelationAttention_15745350107405
// MI455X (gfx1250) — compile-verified
//
#include <hip/hip_runtime.h>
#include <math.h>

// Problem constants (match reference)
constexpr int kB = 16;
constexpr int kL = 256;
constexpr int kH = 768;
constexpr int kR = 64;
constexpr int kRD = 768;   // REL_DIM
constexpr int kA = 256;

typedef __attribute__((ext_vector_type(2))) float v2f;
typedef __attribute__((ext_vector_type(8))) float v8f;

// ---------------------------------------------------------------------------
// f32 WMMA 16x16x4 fragment helpers (CDNA5 layouts, wave32)
//  A (16x4): lanes 0-15 hold M=0..15 {K=0,1}; lanes 16-31 hold {K=2,3}
//  B (4x16): lanes 0-15 hold N=0..15 {K=0,1}; lanes 16-31 hold {K=2,3}
//  C/D (16x16): VGPR i, lanes 0-15 -> M=i, lanes 16-31 -> M=i+8; N = lane&15
// ---------------------------------------------------------------------------
__device__ __forceinline__ v2f load_a_frag(const float* __restrict__ A, int row0,
                                           int k0, int lda, int lane) {
    const int m  = lane & 15;
    const int kk = (lane >> 4) << 1;
    v2f a;
    a.x = A[(row0 + m) * lda + k0 + kk];
    a.y = A[(row0 + m) * lda + k0 + kk + 1];
    return a;
}

__device__ __forceinline__ v2f load_b_frag(const float* __restrict__ Bm, int k0,
                                           int col0, int ldb, int lane) {
    const int n  = lane & 15;
    const int kk = (lane >> 4) << 1;
    v2f b;
    b.x = Bm[(k0 + kk) * ldb + col0 + n];
    b.y = Bm[(k0 + kk + 1) * ldb + col0 + n];
    return b;
}

__device__ __forceinline__ float fast_tanh(float x) {
    float ax = fabsf(x);
    float t  = __expf(-2.0f * ax);          // v_exp_f32 (TRANS pipe)
    float r  = (1.0f - t) * __frcp_rn(1.0f + t);
    return copysignf(r, x);
}

// ---------------------------------------------------------------------------
// Generic C = A(MxK) @ B(KxN) + bias[n], f32 WMMA, one wave per 16x16 tile.
// M,N multiples of 16; K multiple of 4.
// ---------------------------------------------------------------------------
__global__ __launch_bounds__(256)
void wmma_gemm_bias(const float* __restrict__ A, const float* __restrict__ Bm,
                    const float* __restrict__ bias, float* __restrict__ C,
                    int M, int N, int K) {
    const int wave   = blockIdx.x * (blockDim.x >> 5) + (threadIdx.x >> 5);
    const int ntiles = N >> 4;
    const int total  = (M >> 4) * ntiles;
    if (wave >= total) return;
    const int tm = wave / ntiles, tn = wave % ntiles;
    const int row0 = tm << 4, col0 = tn << 4;
    const int lane = threadIdx.x & 31;

    v8f acc = {};
    for (int k = 0; k < K; k += 4) {
        v2f af = load_a_frag(A, row0, k, K, lane);
        v2f bf = load_b_frag(Bm, k, col0, N, lane);
        acc = __builtin_amdgcn_wmma_f32_16x16x4_f32(false, af, false, bf,
                                                    (short)0, acc, false, false);
    }
    const int n     = lane & 15;
    const int rbase = (lane >> 4) << 3;
    const float bc  = bias ? bias[col0 + n] : 0.0f;
#pragma unroll
    for (int i = 0; i < 8; ++i)
        C[(row0 + rbase + i) * N + col0 + n] = acc[i] + bc;
}

// ---------------------------------------------------------------------------
// wxgT[b][a][l] = (X @ Wx)[b*L+l, a] + bx[a] + wg[b, a]
// M = B*L (4096), K = H (768), N = A (256). Output stored transposed per
// batch so the e-kernel reads contiguously along l.
// ---------------------------------------------------------------------------
__global__ __launch_bounds__(256)
void wx_fused_kernel(const float* __restrict__ X, const float* __restrict__ Wx,
                     const float* __restrict__ bx, const float* __restrict__ wg,
                     float* __restrict__ wxgT) {
    const int wave   = blockIdx.x * (blockDim.x >> 5) + (threadIdx.x >> 5);
    const int ntiles = kA >> 4;                       // 16
    const int total  = ((kB * kL) >> 4) * ntiles;     // 4096
    if (wave >= total) return;
    const int tm = wave / ntiles, tn = wave % ntiles;
    const int row0 = tm << 4, col0 = tn << 4;
    const int lane = threadIdx.x & 31;
    const int b    = row0 / kL;                        // tiles never cross b (L%16==0)
    const int l0   = row0 - b * kL;

    v8f acc = {};
    for (int k = 0; k < kH; k += 4) {
        v2f af = load_a_frag(X, row0, k, kH, lane);
        v2f bf = load_b_frag(Wx, k, col0, kA, lane);
        acc = __builtin_amdgcn_wmma_f32_16x16x4_f32(false, af, false, bf,
                                                    (short)0, acc, false, false);
    }
    const int n     = lane & 15;
    const int a_idx = col0 + n;
    const float add = bx[a_idx] + wg[b * kA + a_idx];
    const int rbase = (lane >> 4) << 3;
    float* outb = wxgT + (size_t)b * kA * kL + (size_t)a_idx * kL;
#pragma unroll
    for (int i = 0; i < 8; ++i)
        outb[l0 + rbase + i] = acc[i] + add;
}

// ---------------------------------------------------------------------------
// e[b,r,l] = sum_a tanh(wxgT[b,a,l] + wr[r,a]) * V[a] + bv, masked softmax
// over l. One block (256 thr) per (b,r); thread tid owns l = tid.
// Reads of wxgT are fully coalesced (contiguous in l).
// ---------------------------------------------------------------------------
__global__ __launch_bounds__(256)
void e_softmax_kernel(const float* __restrict__ wxgT, const float* __restrict__ wr,
                      const float* __restrict__ V, const float* __restrict__ bv,
                      const unsigned char* __restrict__ mask,
                      float* __restrict__ aOut) {
    __shared__ float wr_s[kA];
    __shared__ float v_s[kA];
    __shared__ float red[kL];

    const int b = blockIdx.x / kR;
    const int r = blockIdx.x % kR;
    const int tid = threadIdx.x;          // = l

    wr_s[tid] = wr[r * kA + tid];
    v_s[tid]  = V[tid];
    __syncthreads();

    const float* base = wxgT + (size_t)b * kA * kL + tid;
    float acc = 0.0f;
#pragma unroll 4
    for (int a = 0; a < kA; ++a)
        acc += fast_tanh(base[(size_t)a * kL] + wr_s[a]) * v_s[a];

    const bool ok = mask[b * kL + tid] != 0;
    float e = ok ? (acc + bv[0]) : -INFINITY;

    // max reduction
    red[tid] = e;
    __syncthreads();
    for (int s = kL >> 1; s > 0; s >>= 1) {
        if (tid < s) red[tid] = fmaxf(red[tid], red[tid + s]);
        __syncthreads();
    }
    const float mx = red[0];
    __syncthreads();

    const float ex = (ok && mx > -INFINITY) ? __expf(e - mx) : 0.0f;
    red[tid] = ex;
    __syncthreads();
    for (int s = kL >> 1; s > 0; s >>= 1) {
        if (tid < s) red[tid] += red[tid + s];
        __syncthreads();
    }
    const float inv = red[0] > 0.0f ? __frcp_rn(red[0]) : 0.0f;

    aOut[((size_t)b * kR + r) * kL + tid] = ex * inv;
}

// ---------------------------------------------------------------------------
// Batched c[b] = a[b] (RxL) @ X[b] (LxH), f32 WMMA, one wave per 16x16 tile.
// ---------------------------------------------------------------------------
__global__ __launch_bounds__(256)
void c_gemm_kernel(const float* __restrict__ aAtt, const float* __restrict__ X,
                   float* __restrict__ cOut) {
    const int tilesM = kR >> 4;            // 4
    const int tilesN = kH >> 4;            // 48
    const int perB   = tilesM * tilesN;    // 192
    const int wave   = blockIdx.x * (blockDim.x >> 5) + (threadIdx.x >> 5);
    if (wave >= kB * perB) return;
    const int b  = wave / perB;
    const int t  = wave % perB;
    const int tm = t / tilesN, tn = t % tilesN;
    const int row0 = tm << 4, col0 = tn << 4;
    const int lane = threadIdx.x & 31;

    const float* Ab = aAtt + (size_t)b * kR * kL;   // lda = L
    const float* Bb = X + (size_t)b * kL * kH;      // ldb = H

    v8f acc = {};
    for (int k = 0; k < kL; k += 4) {
        v2f af = load_a_frag(Ab, row0, k, kL, lane);
        v2f bf = load_b_frag(Bb, k, col0, kH, lane);
        acc = __builtin_amdgcn_wmma_f32_16x16x4_f32(false, af, false, bf,
                                                    (short)0, acc, false, false);
    }
    const int n     = lane & 15;
    const int rbase = (lane >> 4) << 3;
    float* Cb = cOut + (size_t)b * kR * kH;
#pragma unroll
    for (int i = 0; i < 8; ++i)
        Cb[(row0 + rbase + i) * kH + col0 + n] = acc[i];
}

// ---------------------------------------------------------------------------
extern "C" void kernel_launch(void* const* d_in, const int* in_sizes, int n_in,
                              void* d_out, int out_size, void* d_ws, size_t ws_size,
                              hipStream_t stream) {
    const float* X     = (const float*)d_in[0];
    const float* relE  = (const float*)d_in[1];
    const float* tmean = (const float*)d_in[2];
    const unsigned char* mask = (const unsigned char*)d_in[3];
    const float* Wx = (const float*)d_in[4];
    const float* bx = (const float*)d_in[5];
    const float* Wr = (const float*)d_in[6];
    const float* br = (const float*)d_in[7];
    const float* Wg = (const float*)d_in[8];
    const float* bg = (const float*)d_in[9];
    const float* V  = (const float*)d_in[10];
    const float* bv = (const float*)d_in[11];

    float* ws    = (float*)d_ws;
    float* wg_ws = ws;                                  // kB*kA
    float* wr_ws = wg_ws + kB * kA;                     // kR*kA
    float* wxgT  = wr_ws + kR * kA;                     // kB*kA*kL (4 MB)

    float* cOut = (float*)d_out;                        // (B,R,H)
    float* aOut = cOut + (size_t)kB * kR * kH;          // (B,R,L)

    // 1) wg = tokens_mean @ Wg + bg   (16x768x256)  : 16 tiles
    wmma_gemm_bias<<<2, 256, 0, stream>>>(tmean, Wg, bg, wg_ws, kB, kA, kH);
    // 2) wr = relation_embedding @ Wr + br (64x768x256) : 64 tiles
    wmma_gemm_bias<<<8, 256, 0, stream>>>(relE, Wr, br, wr_ws, kR, kA, kRD);
    // 3) wxgT = transpose(X@Wx + bx + wg) : 4096 tiles, 8 waves/block
    wx_fused_kernel<<<512, 256, 0, stream>>>(X, Wx, bx, wg_ws, wxgT);
    // 4) e + tanh + masked softmax -> a  : one block per (b,r)
    e_softmax_kernel<<<kB * kR, 256, 0, stream>>>(wxgT, wr_ws, V, bv, mask, aOut);
    // 5) c = a @ X (batched) : 3072 tiles, 8 waves/block
    c_gemm_kernel<<<384, 256, 0, stream>>>(aOut, X, cOut);
}